// NCM_classifier_76501957476946
// MI455X (gfx1250) — compile-verified
//
#include <hip/hip_runtime.h>
#include <hip/hip_bf16.h>
#include <math.h>

typedef __attribute__((ext_vector_type(2))) float v2f;
typedef __attribute__((ext_vector_type(8))) float v8f;

#define B_ROWS 1024
#define C_COLS 500
#define FDIM   512
#define N_CTILES 32         // 500 cols padded to 32 tiles of 16
#define LDSX_STRIDE 516     // 512 + 4 pad: lane*516 mod 64 banks distinct for 16 lanes

// ---------------------------------------------------------------------------
// Kernel 1: squared row norms. Rows [0,1024) -> x, rows [1024,1524) -> means.
// ---------------------------------------------------------------------------
__global__ void __launch_bounds__(128)
row_norms_kernel(const float* __restrict__ x,
                 const float* __restrict__ means,
                 float* __restrict__ norms) {
  __shared__ float red[128];
  const int row = blockIdx.x;
  const float* src = (row < B_ROWS) ? (x + (size_t)row * FDIM)
                                    : (means + (size_t)(row - B_ROWS) * FDIM);
  const int t = threadIdx.x;
  float s = 0.f;
#pragma unroll
  for (int i = 0; i < FDIM / 128; ++i) {
    float v = src[t + i * 128];
    s += v * v;
  }
  red[t] = s;
  __syncthreads();
  for (int off = 64; off > 0; off >>= 1) {
    if (t < off) red[t] += red[t + off];
    __syncthreads();
  }
  if (t == 0) norms[row] = red[0];
}

// ---------------------------------------------------------------------------
// Kernel 2: f32-WMMA distance GEMM.
// Grid: (64 row-blocks, 8 col-groups). Block: 128 threads = 4 wave32.
// Each wave computes one 16x16 tile: dist = sqrt(s^2|x|^2 - 2 s x.m + |m|^2).
// ---------------------------------------------------------------------------
__global__ void __launch_bounds__(128)
ncm_wmma_kernel(const float* __restrict__ x,
                const float* __restrict__ means,
                const float* __restrict__ variance,
                const float* __restrict__ norms,
                float* __restrict__ expo_out,
                float* __restrict__ dist_out) {
  __shared__ float sX[16 * LDSX_STRIDE];   // x block, padded stride (33 KB)
  __shared__ float sXN[16];                // s^2 * ||x_row||^2

  const int tid  = threadIdx.x;
  const int wave = tid >> 5;
  const int lane = tid & 31;
  const int row0 = blockIdx.x * 16;

  const float inv  = 1.0f / variance[0];
  const float inv2 = inv * inv;

  if (tid < 16) sXN[tid] = inv2 * norms[row0 + tid];

  // Stage the 16x512 x block into LDS (float4 cooperative copy, 16 per thread).
  for (int i = tid; i < 16 * (FDIM / 4); i += 128) {
    const int r  = i / (FDIM / 4);
    const int c4 = i % (FDIM / 4);
    const float4 v = ((const float4*)(x + (size_t)(row0 + r) * FDIM))[c4];
    float* dst = &sX[r * LDSX_STRIDE + c4 * 4];
    dst[0] = v.x; dst[1] = v.y; dst[2] = v.z; dst[3] = v.w;
  }
  __syncthreads();

  // WMMA fragment addressing (V_WMMA_F32_16X16X4_F32):
  //  A (16x4): lanes 0-15 -> M=lane, K={k,k+1}; lanes 16-31 -> M=lane-16, K={k+2,k+3}
  //  B (4x16): lanes 0-15 -> N=lane, K={k,k+1}; lanes 16-31 -> N=lane-16, K={k+2,k+3}
  const int lhalf = lane >> 4;
  const int l16   = lane & 15;
  const float* aBase = &sX[l16 * LDSX_STRIDE + (lhalf << 1)];

  const int ct   = blockIdx.y * 4 + wave;                  // column tile 0..31
  const int colN = (ct << 4) + l16;                        // this lane's class col
  const int colC = (colN < C_COLS) ? colN : (C_COLS - 1);  // clamp for safe loads
  const float* bBase = means + (size_t)colC * FDIM + (lhalf << 1);

  v8f acc = {};
#pragma unroll 4
  for (int k = 0; k < FDIM; k += 4) {
    v2f a = *(const v2f*)(aBase + k);   // ds_load_b64
    v2f b = *(const v2f*)(bBase + k);   // global_load_b64
    acc = __builtin_amdgcn_wmma_f32_16x16x4_f32(
        /*neg_a=*/false, a, /*neg_b=*/false, b,
        /*c_mod=*/(short)0, acc, /*reuse_a=*/false, /*reuse_b=*/false);
  }

  const float mnorm = norms[B_ROWS + colC];
  const bool  valid = (colN < C_COLS);
#pragma unroll
  for (int v = 0; v < 8; ++v) {
    const int m = v + (lhalf << 3);          // C/D layout: M = v + 8*half, N = l16
    const float g  = acc[v];                 // x . mean (unscaled)
    float d2 = sXN[m] - 2.0f * inv * g + mnorm;
    d2 = fmaxf(d2, 0.0f);
    const float d = sqrtf(d2);
    if (valid) {
      const size_t oi = (size_t)(row0 + m) * C_COLS + colN;
      dist_out[oi] = d;
      expo_out[oi] = -d;
    }
  }
}

// ---------------------------------------------------------------------------
// Kernel 3: row softmax over 500 classes. One block (128 threads) per row.
// ---------------------------------------------------------------------------
__global__ void __launch_bounds__(128)
softmax_kernel(const float* __restrict__ expo,
               float* __restrict__ probs) {
  __shared__ float red[128];
  const int row = blockIdx.x;
  const int t   = threadIdx.x;
  const float* e = expo + (size_t)row * C_COLS;
  float* p       = probs + (size_t)row * C_COLS;

  float m = -INFINITY;
  for (int c = t; c < C_COLS; c += 128) m = fmaxf(m, e[c]);
  red[t] = m;
  __syncthreads();
  for (int off = 64; off > 0; off >>= 1) {
    if (t < off) red[t] = fmaxf(red[t], red[t + off]);
    __syncthreads();
  }
  const float rmax = red[0];
  __syncthreads();

  float s = 0.f;
  for (int c = t; c < C_COLS; c += 128) s += __expf(e[c] - rmax);
  red[t] = s;
  __syncthreads();
  for (int off = 64; off > 0; off >>= 1) {
    if (t < off) red[t] += red[t + off];
    __syncthreads();
  }
  const float rinv = 1.0f / red[0];

  for (int c = t; c < C_COLS; c += 128) p[c] = __expf(e[c] - rmax) * rinv;
}

// ---------------------------------------------------------------------------
extern "C" void kernel_launch(void* const* d_in, const int* in_sizes, int n_in,
                              void* d_out, int out_size, void* d_ws, size_t ws_size,
                              hipStream_t stream) {
  const float* x        = (const float*)d_in[0];  // [1024, 512]
  const float* means    = (const float*)d_in[1];  // [500, 512]
  const float* variance = (const float*)d_in[2];  // [1]

  float* out   = (float*)d_out;
  float* probs = out;                                   // [1024*500]
  float* expo  = out + (size_t)B_ROWS * C_COLS;         // [1024*500]
  float* dist  = out + 2 * (size_t)B_ROWS * C_COLS;     // [1024*500]

  float* norms = (float*)d_ws;                          // 1524 floats of scratch

  row_norms_kernel<<<B_ROWS + C_COLS, 128, 0, stream>>>(x, means, norms);

  dim3 grid(B_ROWS / 16, N_CTILES / 4);                 // 64 x 8 = 512 workgroups
  ncm_wmma_kernel<<<grid, 128, 0, stream>>>(x, means, variance, norms,
                                            expo, dist);

  softmax_kernel<<<B_ROWS, 128, 0, stream>>>(expo, probs);
}